// TopKGate_62508954026516
// MI455X (gfx1250) — compile-verified
//
#include <hip/hip_runtime.h>
#include <hip/hip_bf16.h>
#include <stdint.h>

typedef float v2f __attribute__((ext_vector_type(2)));
typedef float v4f __attribute__((ext_vector_type(4)));
typedef float v8f __attribute__((ext_vector_type(8)));

#define T_TOKENS 8192
#define H_DIM    2048
#define E_EXP    64
#define CAPC     256            // C = TOPK * cap = 2 * 128
#define EPS_F    1.1920929e-07f

// ---------------------------------------------------------------------------
// Kernel 0: zero the gsum accumulator (ws is poisoned 0xAA by the harness)
// ---------------------------------------------------------------------------
__global__ void init_ws_kernel(float* gsum) {
    if (threadIdx.x < E_EXP) gsum[threadIdx.x] = 0.0f;
}

// ---------------------------------------------------------------------------
// Kernel 1: fused GEMM (WMMA fp32 16x16x4) + softmax + top-2 per token.
// Block = 256 threads (8 waves). Each wave computes a 16-token x 64-expert
// logit tile; block handles 128 tokens. 64 blocks cover T=8192.
// ---------------------------------------------------------------------------
__global__ __launch_bounds__(256)
void gate_gemm_softmax_kernel(const float* __restrict__ x,
                              const float* __restrict__ W,
                              int* __restrict__ e1, int* __restrict__ e2,
                              float* __restrict__ w1o, float* __restrict__ w2o,
                              float* __restrict__ gsum)
{
    __shared__ float lds[128 * 65];   // padded stride 65 to dodge bank conflicts
    __shared__ float tmax[128];
    __shared__ float tden[128];

    const int tid  = threadIdx.x;
    const int lane = tid & 31;
    const int wv   = tid >> 5;
    const int m16  = lane & 15;       // row-within-tile / col-within-tile
    const int h    = lane >> 4;       // half-wave select
    const int rowBase = blockIdx.x * 128 + wv * 16;

    const float* xrow = x + (size_t)(rowBase + m16) * H_DIM;
    const float* wr0  = W + (size_t)( 0 + m16) * H_DIM;
    const float* wr1  = W + (size_t)(16 + m16) * H_DIM;
    const float* wr2  = W + (size_t)(32 + m16) * H_DIM;
    const float* wr3  = W + (size_t)(48 + m16) * H_DIM;

    v8f acc0 = {}; v8f acc1 = {}; v8f acc2 = {}; v8f acc3 = {};

    // K loop: each WMMA consumes K=4.  A lane holds A[m16][kk+2h+{0,1}] and
    // B[kk+2h+{0,1}][n16] = W[n16][kk+2h+{0,1}] -> contiguous float2 loads.
    for (int kk = 0; kk < H_DIM; kk += 4) {
        const int ko = kk + 2 * h;
        v2f a  = *(const v2f*)(xrow + ko);
        v2f b0 = *(const v2f*)(wr0 + ko);
        v2f b1 = *(const v2f*)(wr1 + ko);
        v2f b2 = *(const v2f*)(wr2 + ko);
        v2f b3 = *(const v2f*)(wr3 + ko);
        acc0 = __builtin_amdgcn_wmma_f32_16x16x4_f32(false, a, false, b0, (short)0, acc0, false, false);
        acc1 = __builtin_amdgcn_wmma_f32_16x16x4_f32(false, a, false, b1, (short)0, acc1, false, false);
        acc2 = __builtin_amdgcn_wmma_f32_16x16x4_f32(false, a, false, b2, (short)0, acc2, false, false);
        acc3 = __builtin_amdgcn_wmma_f32_16x16x4_f32(false, a, false, b3, (short)0, acc3, false, false);
    }

    // Spill logits to LDS. C/D layout: VGPR r -> token m = r + 8*h, col = m16.
#pragma unroll
    for (int r = 0; r < 8; ++r) {
        const int tok = wv * 16 + r + 8 * h;
        float* dst = &lds[tok * 65];
        dst[ 0 + m16] = acc0[r];
        dst[16 + m16] = acc1[r];
        dst[32 + m16] = acc2[r];
        dst[48 + m16] = acc3[r];
    }
    __syncthreads();

    // Phase 2a: one thread per token (threads 0..127): softmax stats + top-2.
    if (tid < 128) {
        const float* row = &lds[tid * 65];
        float mx = row[0];
#pragma unroll
        for (int e = 1; e < E_EXP; ++e) mx = fmaxf(mx, row[e]);
        float den = 0.0f;
        float b1v = -3.4e38f; int b1i = 0;
        float b2v = -3.4e38f; int b2i = 0;
#pragma unroll
        for (int e = 0; e < E_EXP; ++e) {
            float l = row[e];
            den += __expf(l - mx);
            if (l > b1v)      { b2v = b1v; b2i = b1i; b1v = l; b1i = e; }
            else if (l > b2v) { b2v = l;   b2i = e; }
        }
        const float g1 = __expf(b1v - mx) / den;
        const float g2 = __expf(b2v - mx) / den;
        const float gs = fmaxf(g1 + g2, EPS_F);
        const int tok = blockIdx.x * 128 + tid;
        e1[tok]  = b1i;
        e2[tok]  = b2i;
        w1o[tok] = g1 / gs;
        w2o[tok] = g2 / gs;
        tmax[tid] = mx;
        tden[tid] = den;
    }
    __syncthreads();

    // Phase 2b: per-expert partial gate sums (for l_aux), threads 0..63.
    if (tid < E_EXP) {
        float p = 0.0f;
        for (int t = 0; t < 128; ++t)
            p += __expf(lds[t * 65 + tid] - tmax[t]) / tden[t];
        atomicAdd(&gsum[tid], p);
    }
}

// ---------------------------------------------------------------------------
// Kernel 2: per-expert stable scan -> capacity positions + expert counts.
// One block per expert (64 blocks x 256 threads), ballot-based block scan.
// Ordering matches reference: all slot-0 picks (t ascending), then slot-1.
// ---------------------------------------------------------------------------
__global__ __launch_bounds__(256)
void expert_scan_kernel(const int* __restrict__ e1, const int* __restrict__ e2,
                        int* __restrict__ pos0, int* __restrict__ pos1,
                        int* __restrict__ cnt, float* __restrict__ exp_counts_out)
{
    __shared__ int wsum[8];
    const int e    = blockIdx.x;
    const int tid  = threadIdx.x;
    const int lane = tid & 31;
    const int wv   = tid >> 5;
    const uint64_t ltmask = (1ull << lane) - 1ull;

    int base = 0;
    // slot 0
    for (int t0 = 0; t0 < T_TOKENS; t0 += 256) {
        const int t = t0 + tid;
        const bool flag = (e1[t] == e);
        const uint64_t bal = __ballot(flag);
        if (lane == 0) wsum[wv] = __popcll(bal);
        __syncthreads();
        int prefix = 0, total = 0;
#pragma unroll
        for (int i = 0; i < 8; ++i) { int v = wsum[i]; if (i < wv) prefix += v; total += v; }
        if (flag) pos0[t] = base + prefix + __popcll(bal & ltmask);
        base += total;
        __syncthreads();
    }
    // slot 1 (continues the running count: c0_total + prefix within slot1)
    for (int t0 = 0; t0 < T_TOKENS; t0 += 256) {
        const int t = t0 + tid;
        const bool flag = (e2[t] == e);
        const uint64_t bal = __ballot(flag);
        if (lane == 0) wsum[wv] = __popcll(bal);
        __syncthreads();
        int prefix = 0, total = 0;
#pragma unroll
        for (int i = 0; i < 8; ++i) { int v = wsum[i]; if (i < wv) prefix += v; total += v; }
        if (flag) pos1[t] = base + prefix + __popcll(bal & ltmask);
        base += total;
        __syncthreads();
    }
    if (tid == 0) {
        cnt[e] = base;
        exp_counts_out[e] = (float)base;   // exp_counts in output tuple
    }
}

// ---------------------------------------------------------------------------
// Kernel 3: zero-fill the 1 GiB combine+dispatch region with non-temporal
// B128 stores (region is 5x the 192MB L2 and written once -> TH=NT).
// dst starts at d_out+3 (12B misaligned) -> scalar head/tail, v4f body.
// ---------------------------------------------------------------------------
__global__ __launch_bounds__(256)
void zero_fill_kernel(float* __restrict__ dst, size_t n)
{
    const size_t stride = (size_t)gridDim.x * blockDim.x;
    const size_t tid = (size_t)blockIdx.x * blockDim.x + threadIdx.x;
    size_t head = ((uintptr_t)dst & 15) ? ((16 - ((uintptr_t)dst & 15)) >> 2) : 0;
    if (head > n) head = n;
    if (tid < head) __builtin_nontemporal_store(0.0f, dst + tid);

    v4f* d4 = (v4f*)(dst + head);
    const size_t n4 = (n - head) >> 2;
    const v4f z = {0.0f, 0.0f, 0.0f, 0.0f};
    for (size_t i = tid; i < n4; i += stride)
        __builtin_nontemporal_store(z, d4 + i);

    const size_t tail_start = head + (n4 << 2);
    const size_t tail = n - tail_start;
    if (tid < tail) __builtin_nontemporal_store(0.0f, dst + tail_start + tid);
}

// ---------------------------------------------------------------------------
// Kernel 4: scatter the (at most) 2 nonzeros per token into combine/dispatch.
// ---------------------------------------------------------------------------
__global__ __launch_bounds__(256)
void scatter_kernel(const int* __restrict__ e1, const int* __restrict__ e2,
                    const float* __restrict__ w1, const float* __restrict__ w2,
                    const int* __restrict__ pos0, const int* __restrict__ pos1,
                    float* __restrict__ out)
{
    const int t = blockIdx.x * blockDim.x + threadIdx.x;
    if (t >= T_TOKENS) return;
    const size_t TEC = (size_t)T_TOKENS * E_EXP * CAPC;
    float* combine  = out + 3;
    float* dispatch = out + 3 + TEC;
    const size_t baseT = (size_t)t * E_EXP * CAPC;

    const int a = e1[t]; const int p = pos0[t];
    if (p < CAPC) {
        const size_t idx = baseT + (size_t)a * CAPC + (size_t)p;
        __builtin_nontemporal_store(w1[t], combine + idx);
        __builtin_nontemporal_store(1.0f,  dispatch + idx);
    }
    const int b = e2[t]; const int q = pos1[t];
    if (q < CAPC) {
        const size_t idx = baseT + (size_t)b * CAPC + (size_t)q;
        __builtin_nontemporal_store(w2[t], combine + idx);
        __builtin_nontemporal_store(1.0f,  dispatch + idx);
    }
}

// ---------------------------------------------------------------------------
// Kernel 5: scalars.  l_aux = E * sum_e (cnt[e]/T)*(gsum[e]/T); z_loss = 0;
// rate = sum_e min(cnt[e], C) / (2T).
// ---------------------------------------------------------------------------
__global__ void finalize_kernel(const float* __restrict__ gsum,
                                const int* __restrict__ cnt,
                                float* __restrict__ out)
{
    __shared__ float s1[E_EXP];
    __shared__ float s2[E_EXP];
    const int e = threadIdx.x;
    if (e < E_EXP) {
        const float me = (float)cnt[e] / (float)T_TOKENS;
        const float ge = gsum[e] / (float)T_TOKENS;
        s1[e] = me * ge;
        s2[e] = (float)(cnt[e] < CAPC ? cnt[e] : CAPC);
    }
    __syncthreads();
    if (e == 0) {
        float l = 0.0f, r = 0.0f;
        for (int i = 0; i < E_EXP; ++i) { l += s1[i]; r += s2[i]; }
        out[0] = (float)E_EXP * l;                   // l_aux
        out[1] = 0.0f;                               // z_loss
        out[2] = r / (float)(2 * T_TOKENS);          // exp_capacity_rate
    }
}

// ---------------------------------------------------------------------------
extern "C" void kernel_launch(void* const* d_in, const int* in_sizes, int n_in,
                              void* d_out, int out_size, void* d_ws, size_t ws_size,
                              hipStream_t stream) {
    const float* x = (const float*)d_in[0];   // [4,2048,2048] -> [8192,2048]
    const float* W = (const float*)d_in[1];   // [64,2048]
    float* out = (float*)d_out;

    // workspace layout
    int*   e1    = (int*)d_ws;
    int*   e2    = e1 + T_TOKENS;
    float* w1    = (float*)(e2 + T_TOKENS);
    float* w2    = w1 + T_TOKENS;
    int*   pos0  = (int*)(w2 + T_TOKENS);
    int*   pos1  = pos0 + T_TOKENS;
    float* gsum  = (float*)(pos1 + T_TOKENS);
    int*   cnt   = (int*)(gsum + E_EXP);

    const size_t TEC = (size_t)T_TOKENS * E_EXP * CAPC;
    float* exp_counts_out = out + 3 + 2 * TEC;

    // 1) init accumulators
    init_ws_kernel<<<1, 64, 0, stream>>>(gsum);

    // 2) fused WMMA GEMM + softmax + top2   (64 blocks * 128 tokens)
    gate_gemm_softmax_kernel<<<T_TOKENS / 128, 256, 0, stream>>>(
        x, W, e1, e2, w1, w2, gsum);

    // 3) per-expert capacity scan
    expert_scan_kernel<<<E_EXP, 256, 0, stream>>>(e1, e2, pos0, pos1, cnt,
                                                  exp_counts_out);

    // 4) zero-fill combine + dispatch (1 GiB, NT stores)
    zero_fill_kernel<<<2048, 256, 0, stream>>>(out + 3, 2 * TEC);

    // 5) scatter nonzeros
    scatter_kernel<<<(T_TOKENS + 255) / 256, 256, 0, stream>>>(
        e1, e2, w1, w2, pos0, pos1, out);

    // 6) scalars
    finalize_kernel<<<1, 64, 0, stream>>>(gsum, cnt, out);
}